// GraphAutoencoder_26568667693118
// MI455X (gfx1250) — compile-verified
//
#include <hip/hip_runtime.h>

// ---------------------------------------------------------------------------
// GCN autoencoder (DOMINANT-style) for gfx1250.
//   N=10000, IN=1024, HID=128, LAT=64, E=320000
// Dense matmuls + s@s^T use V_WMMA_F32_16X16X4_F32 (full f32 matrix path).
// All GEMM shapes are compile-time (templates) so B-row strides fold into
// immediate offsets. The self-loop scale (dis^2 * h) is fused into the GEMM
// epilogue. Graph propagation uses global_atomic_add_f32 into L2-resident
// buffers (everything except struct_recon fits in the 192MB L2).
// ---------------------------------------------------------------------------

typedef float v2f __attribute__((ext_vector_type(2)));
typedef float v8f __attribute__((ext_vector_type(8)));

#define NN      10000
#define IN_DIM  1024
#define HID_DIM 128
#define LAT_DIM 64
#define NE      320000

// ---------------------------------------------------------------------------
// Degree / normalization
// ---------------------------------------------------------------------------
__global__ void k_init_deg(float* __restrict__ deg) {
    int i = blockIdx.x * blockDim.x + threadIdx.x;
    if (i < NN) deg[i] = 1.0f;                 // self-loop contributes 1
}

__global__ void k_accum_deg(const long long* __restrict__ ei,
                            float* __restrict__ deg) {
    int e = blockIdx.x * blockDim.x + threadIdx.x;
    if (e < NE) {
        int r = (int)ei[NE + e];               // recv = edge_index[1]
        atomicAdd(&deg[r], 1.0f);
    }
}

__global__ void k_finalize_dis(float* __restrict__ dis) {
    int i = blockIdx.x * blockDim.x + threadIdx.x;
    if (i < NN) {
        float d = dis[i];
        dis[i] = (d > 0.0f) ? rsqrtf(d) : 0.0f;
    }
}

// ---------------------------------------------------------------------------
// GEMM (row-major A[M,K] x row-major B[K,Nn] -> C[M,Nn]) via f32 WMMA.
// One wave computes a 16x64 C tile (4 accumulators); block = 4 waves stacked
// in M. Requires 16|M (10000=625*16) and 64|Nn. Nn,K compile-time so all
// strides become immediate offsets. Epilogue also writes agg = dis[row]^2 * C
// (the GCN self-loop term), saving a full re-read of C.
// ---------------------------------------------------------------------------
template <int Nn, int K>
__global__ __launch_bounds__(128) void k_gemm_rr(const float* __restrict__ A,
                                                 const float* __restrict__ B,
                                                 float* __restrict__ C,
                                                 float* __restrict__ agg,
                                                 const float* __restrict__ dis,
                                                 int M) {
    const int lane = threadIdx.x & 31;
    const int wave = threadIdx.x >> 5;
    const int l2   = lane & 15;
    const int kh   = (lane >> 4) * 2;          // f32 frag: lanes 0-15 K={0,1}, 16-31 K={2,3}
    const int mtile = blockIdx.y * 4 + wave;
    if (mtile * 16 >= M) return;               // uniform per wave: EXEC stays all-1s
    const int m0 = mtile * 16;
    const int n0 = blockIdx.x * 64;

    v8f acc0 = {}, acc1 = {}, acc2 = {}, acc3 = {};
    const float* arow = A + (size_t)(m0 + l2) * K + kh;
    const float* bbase = B + (size_t)kh * Nn + n0 + l2;

#pragma unroll 4
    for (int k = 0; k < K; k += 4) {
        v2f af;
        af.x = arow[k];
        af.y = arow[k + 1];
        const float* bp = bbase + (size_t)k * Nn;
        v2f b0, b1, b2, b3;
        b0.x = bp[0];  b0.y = bp[Nn];
        b1.x = bp[16]; b1.y = bp[Nn + 16];
        b2.x = bp[32]; b2.y = bp[Nn + 32];
        b3.x = bp[48]; b3.y = bp[Nn + 48];
        acc0 = __builtin_amdgcn_wmma_f32_16x16x4_f32(false, af, false, b0, (short)0, acc0, false, false);
        acc1 = __builtin_amdgcn_wmma_f32_16x16x4_f32(false, af, false, b1, (short)0, acc1, false, false);
        acc2 = __builtin_amdgcn_wmma_f32_16x16x4_f32(false, af, false, b2, (short)0, acc2, false, false);
        acc3 = __builtin_amdgcn_wmma_f32_16x16x4_f32(false, af, false, b3, (short)0, acc3, false, false);
    }

    const int row0 = m0 + (lane >> 4) * 8;     // C/D layout: VGPR r -> rows r and r+8
#pragma unroll
    for (int r = 0; r < 8; ++r) {
        const int row = row0 + r;
        float dd = dis[row];
        dd *= dd;
        float* crow = C   + (size_t)row * Nn + n0 + l2;
        float* grow = agg + (size_t)row * Nn + n0 + l2;
        crow[0]  = acc0[r];      grow[0]  = dd * acc0[r];
        crow[16] = acc1[r];      grow[16] = dd * acc1[r];
        crow[32] = acc2[r];      grow[32] = dd * acc2[r];
        crow[48] = acc3[r];      grow[48] = dd * acc3[r];
    }
}

// ---------------------------------------------------------------------------
// C = A * Bt^T  (A[NN,K], Bt[NN,K] row-major)  -- struct_recon = s @ s^T.
// NN=10000 is not a multiple of 64: trailing 16-wide subtiles masked with
// uniform branches (EXEC stays all-1s through every executed WMMA). K=64 is
// compile-time so the K loop fully unrolls with immediate-offset b64 loads.
// ---------------------------------------------------------------------------
template <int K>
__global__ __launch_bounds__(128) void k_gemm_nt(const float* __restrict__ A,
                                                 const float* __restrict__ Bt,
                                                 float* __restrict__ C) {
    const int lane = threadIdx.x & 31;
    const int wave = threadIdx.x >> 5;
    const int l2   = lane & 15;
    const int kh   = (lane >> 4) * 2;
    const int mtile = blockIdx.y * 4 + wave;
    if (mtile * 16 >= NN) return;
    const int m0 = mtile * 16;
    const int n0 = blockIdx.x * 64;
    int nsub = (NN - n0 + 15) >> 4;            // valid 16-col subtiles (16|NN)
    nsub = nsub > 4 ? 4 : nsub;

    v8f acc0 = {}, acc1 = {}, acc2 = {}, acc3 = {};
    const float* arow = A + (size_t)(m0 + l2) * K + kh;
    const float* b0r = Bt + (size_t)(n0 + l2) * K + kh;
    const float* b1r = (nsub > 1) ? Bt + (size_t)(n0 + 16 + l2) * K + kh : b0r;
    const float* b2r = (nsub > 2) ? Bt + (size_t)(n0 + 32 + l2) * K + kh : b0r;
    const float* b3r = (nsub > 3) ? Bt + (size_t)(n0 + 48 + l2) * K + kh : b0r;

#pragma unroll
    for (int k = 0; k < K; k += 4) {
        v2f af, bf;
        af.x = arow[k];
        af.y = arow[k + 1];
        bf.x = b0r[k]; bf.y = b0r[k + 1];
        acc0 = __builtin_amdgcn_wmma_f32_16x16x4_f32(false, af, false, bf, (short)0, acc0, false, false);
        if (nsub > 1) {
            bf.x = b1r[k]; bf.y = b1r[k + 1];
            acc1 = __builtin_amdgcn_wmma_f32_16x16x4_f32(false, af, false, bf, (short)0, acc1, false, false);
        }
        if (nsub > 2) {
            bf.x = b2r[k]; bf.y = b2r[k + 1];
            acc2 = __builtin_amdgcn_wmma_f32_16x16x4_f32(false, af, false, bf, (short)0, acc2, false, false);
        }
        if (nsub > 3) {
            bf.x = b3r[k]; bf.y = b3r[k + 1];
            acc3 = __builtin_amdgcn_wmma_f32_16x16x4_f32(false, af, false, bf, (short)0, acc3, false, false);
        }
    }

    const int row0 = m0 + (lane >> 4) * 8;
#pragma unroll
    for (int r = 0; r < 8; ++r) {
        float* crow = C + (size_t)(row0 + r) * NN + n0 + l2;
        crow[0] = acc0[r];
        if (nsub > 1) crow[16] = acc1[r];
        if (nsub > 2) crow[32] = acc2[r];
        if (nsub > 3) crow[48] = acc3[r];
    }
}

// ---------------------------------------------------------------------------
// Edge scatter: agg[recv] += dis[send]*dis[recv] * h[send]   (float4 chunks)
// ---------------------------------------------------------------------------
template <int F, int L2F4>
__global__ void k_scatter(const float* __restrict__ h,
                          const long long* __restrict__ ei,
                          const float* __restrict__ dis,
                          float* __restrict__ agg) {
    const int f4m  = (1 << L2F4) - 1;
    const int fsub = threadIdx.x & f4m;        // which float4 chunk
    const int esub = threadIdx.x >> L2F4;
    const int e = blockIdx.x * (256 >> L2F4) + esub;
    if (e >= NE) return;
    const int s = (int)ei[e];
    const int r = (int)ei[NE + e];
    const float w = dis[s] * dis[r];
    const int f = fsub << 2;
    const float4 hv = *(const float4*)(h + (size_t)s * F + f);
    float* ap = agg + (size_t)r * F + f;
    atomicAdd(ap + 0, w * hv.x);
    atomicAdd(ap + 1, w * hv.y);
    atomicAdd(ap + 2, w * hv.z);
    atomicAdd(ap + 3, w * hv.w);
}

// ---------------------------------------------------------------------------
// Fused bias + ReLU, float4-vectorized (16B-aligned buffers, 4|F).
// ---------------------------------------------------------------------------
__global__ void k_bias_relu4(float* __restrict__ agg,
                             const float* __restrict__ b,
                             int total4, int f4mask) {
    int i = blockIdx.x * blockDim.x + threadIdx.x;
    if (i < total4) {
        float4 v = ((const float4*)agg)[i];
        float4 bb = ((const float4*)b)[i & f4mask];
        v.x = fmaxf(v.x + bb.x, 0.0f);
        v.y = fmaxf(v.y + bb.y, 0.0f);
        v.z = fmaxf(v.z + bb.z, 0.0f);
        v.w = fmaxf(v.w + bb.w, 0.0f);
        ((float4*)agg)[i] = v;
    }
}

// ---------------------------------------------------------------------------
// Launch
// ---------------------------------------------------------------------------
static const int MG = (NN / 16 + 3) / 4;       // 157 M-tile groups (4 waves/block)

template <int K, int F, int L2F4>
static void gcn_layer(const float* xin, const float* W, const float* b,
                      float* mm, float* agg, const float* dis,
                      const long long* ei, hipStream_t stream) {
    dim3 g(F / 64, MG);
    k_gemm_rr<F, K><<<g, 128, 0, stream>>>(xin, W, mm, agg, dis, NN);
    const int epb = 256 >> L2F4;               // edges per 256-thread block
    k_scatter<F, L2F4><<<(NE + epb - 1) / epb, 256, 0, stream>>>(mm, ei, dis, agg);
    const int total4 = NN * F / 4;
    k_bias_relu4<<<(total4 + 255) / 256, 256, 0, stream>>>(agg, b, total4, (F >> 2) - 1);
}

extern "C" void kernel_launch(void* const* d_in, const int* in_sizes, int n_in,
                              void* d_out, int out_size, void* d_ws, size_t ws_size,
                              hipStream_t stream) {
    const float*     x      = (const float*)d_in[0];
    const long long* ei     = (const long long*)d_in[1];
    const float*     W_enc1 = (const float*)d_in[2];
    const float*     b_enc1 = (const float*)d_in[3];
    const float*     W_enc2 = (const float*)d_in[4];
    const float*     b_enc2 = (const float*)d_in[5];
    const float*     W_att1 = (const float*)d_in[6];
    const float*     b_att1 = (const float*)d_in[7];
    const float*     W_att2 = (const float*)d_in[8];
    const float*     b_att2 = (const float*)d_in[9];
    const float*     W_str  = (const float*)d_in[10];
    const float*     b_str  = (const float*)d_in[11];

    float* out_attr   = (float*)d_out;                          // [N,1024]
    float* out_struct = out_attr + (size_t)NN * IN_DIM;         // [N,N]
    float* out_z      = out_struct + (size_t)NN * NN;           // [N,64]

    float* ws  = (float*)d_ws;
    float* dis = ws;                                            // [N] (deg then rsqrt)
    float* mm  = ws + 16384;                                    // [N,1024] matmul scratch
    float* hb  = mm + (size_t)NN * 1024;                        // [N,128]
    float* ab  = hb + (size_t)NN * HID_DIM;                     // [N,128]
    float* sb  = ab + (size_t)NN * HID_DIM;                     // [N,64]

    // --- normalization ---
    k_init_deg<<<(NN + 255) / 256, 256, 0, stream>>>(dis);
    k_accum_deg<<<(NE + 255) / 256, 256, 0, stream>>>(ei, dis);
    k_finalize_dis<<<(NN + 255) / 256, 256, 0, stream>>>(dis);

    // --- GCN stack ---
    gcn_layer<IN_DIM,  HID_DIM, 5>(x,     W_enc1, b_enc1, mm, hb,       dis, ei, stream); // h
    gcn_layer<HID_DIM, LAT_DIM, 4>(hb,    W_enc2, b_enc2, mm, out_z,    dis, ei, stream); // z
    gcn_layer<LAT_DIM, HID_DIM, 5>(out_z, W_att1, b_att1, mm, ab,       dis, ei, stream); // a
    gcn_layer<HID_DIM, IN_DIM,  8>(ab,    W_att2, b_att2, mm, out_attr, dis, ei, stream); // attr
    gcn_layer<LAT_DIM, LAT_DIM, 4>(out_z, W_str,  b_str,  mm, sb,       dis, ei, stream); // s

    // --- struct_recon = s @ s^T ---
    dim3 gs((NN + 63) / 64, MG);
    k_gemm_nt<LAT_DIM><<<gs, 128, 0, stream>>>(sb, sb, out_struct);
}